// TripletMPNN_73572789781128
// MI455X (gfx1250) — compile-verified
//
#include <hip/hip_runtime.h>

// ---------------------------------------------------------------------------
// TripletMPNN for MI455X (gfx1250, wave32). fp32 WMMA (V_WMMA_F32_16X16X4_F32)
// for all large GEMMs; problem is HBM/L2 bound (~300MB traffic vs ~7 GFLOP),
// so exact fp32 matrix math is the right choice — no precision downgrade.
// GEMM v2: W panel staged (transposed) in LDS once per block; each wave
// computes a 16 x Nc stripe (up to 8 WMMA accumulators) so every A fragment
// feeds Nc/16 WMMAs. Async global->LDS staging when the toolchain has it.
// ---------------------------------------------------------------------------

#define LN_EPS 1e-5f
#define NEGINF (-1.0e9f)

typedef float v2f __attribute__((ext_vector_type(2)));
typedef float v8f __attribute__((ext_vector_type(8)));

#if defined(__AMDGCN__) && __has_builtin(__builtin_amdgcn_global_load_async_to_lds_b32)
#define HAVE_ASYNC_LDS 1
typedef __attribute__((address_space(1))) int gas_int;   // global b32
typedef __attribute__((address_space(3))) int las_int;   // LDS b32
#endif

__device__ __forceinline__ float wave_sum32(float v) {
#pragma unroll
  for (int off = 16; off > 0; off >>= 1) v += __shfl_xor(v, off, 32);
  return v;
}

// ---------------------------------------------------------------------------
// fp32 WMMA GEMM: C[M, NT*16] = A[M,K] @ W[K, NT*16]
// Block = 128 threads (4 waves); wave w computes rows [ (blk*4+w)*16 , +16 )
// across the full Nc = NT*16 columns (NT v8f accumulators).
// W staged transposed in LDS as ldsW[n][k] so each B fragment is one b64 load.
// VGPR layouts per CDNA5 ISA §7.12.2:
//   A 16x4 : lane l holds row m=l&15; (l>>4) selects K pair {0,1} vs {2,3}
//   B 4x16 : VGPR v, lanes0-15 -> K=v, lanes16-31 -> K=v+2, col = l&15
//   C 16x16: VGPR v, lanes0-15 -> M=v, lanes16-31 -> M=v+8, col = l&15
// Requires M%64==0, K%4==0.
// ---------------------------------------------------------------------------
template <int NT>
__global__ void wmma_gemm_lds(const float* __restrict__ A, const float* __restrict__ W,
                              float* __restrict__ C, int M, int K) {
  extern __shared__ float ldsW[];           // [NT*16][K] (transposed W)
  const int Nc  = NT * 16;
  const int tid = threadIdx.x;              // 0..127
  const int total = K * Nc;                 // multiple of 128 for all our shapes

  // ---- stage W (transposed) into LDS, cooperatively -----------------------
#if defined(HAVE_ASYNC_LDS)
  for (int idx = tid; idx < total; idx += 128) {
    const int k = idx / Nc, n = idx - k * Nc;
    __builtin_amdgcn_global_load_async_to_lds_b32(
        (gas_int*)(uintptr_t)(W + idx),
        (las_int*)(uint32_t)(uintptr_t)(&ldsW[n * K + k]), 0, 0);
  }
#if __has_builtin(__builtin_amdgcn_s_wait_asynccnt)
  __builtin_amdgcn_s_wait_asynccnt(0);
#else
  asm volatile("s_wait_asynccnt 0x0" ::: "memory");
#endif
#else
  for (int idx = tid; idx < total; idx += 128) {
    const int k = idx / Nc, n = idx - k * Nc;
    ldsW[n * K + k] = W[idx];
  }
#endif
  __syncthreads();

  const int lane  = tid & 31;
  const int wave  = tid >> 5;
  const int mbase = (blockIdx.x * 4 + wave) * 16;
  const int half  = lane >> 4;
  const int l16   = lane & 15;

  const float* __restrict__ arow = A + (size_t)(mbase + l16) * K;

  v8f acc[NT];
#pragma unroll
  for (int nt = 0; nt < NT; ++nt) acc[nt] = (v8f){0.f, 0.f, 0.f, 0.f, 0.f, 0.f, 0.f, 0.f};

  for (int k = 0; k < K; k += 4) {
    if ((k & 63) == 0) __builtin_prefetch(arow + k + 64, 0, 1);  // global_prefetch_b8
    const int ks = k + 2 * half;
    const v2f a = *(const v2f*)(arow + ks);          // A[m][ks], A[m][ks+1]
#pragma unroll
    for (int nt = 0; nt < NT; ++nt) {
      const v2f b = *(const v2f*)(&ldsW[(nt * 16 + l16) * K + ks]);  // ds_load_b64
      acc[nt] = __builtin_amdgcn_wmma_f32_16x16x4_f32(false, a, false, b, (short)0,
                                                      acc[nt], false, false);
    }
  }

  float* __restrict__ crow = C + (size_t)(mbase + 8 * half) * Nc + l16;
#pragma unroll
  for (int nt = 0; nt < NT; ++nt) {
#pragma unroll
    for (int v = 0; v < 8; ++v) crow[(size_t)v * Nc + nt * 16] = acc[nt][v];
  }
}

// ---------------------------------------------------------------------------
// Small VALU projections
// ---------------------------------------------------------------------------

// t{1,2,3}[row,c] = mask[row] ? z[row,:] @ Wt{1,2,3}[:,c] : 0   (512 rows, 8 cols)
__global__ void z_proj_kernel(const float* __restrict__ z,
                              const float* __restrict__ Wt1, const float* __restrict__ Wt2,
                              const float* __restrict__ Wt3,
                              const unsigned char* __restrict__ mask,
                              float* __restrict__ t1, float* __restrict__ t2,
                              float* __restrict__ t3) {
  const int t = blockIdx.x * blockDim.x + threadIdx.x;
  if (t >= 512 * 8) return;
  const int row = t >> 3, c = t & 7;
  const float* zr = z + (size_t)row * 256;
  float s1 = 0.f, s2 = 0.f, s3 = 0.f;
  for (int k = 0; k < 256; ++k) {
    const float zv = zr[k];
    s1 += zv * Wt1[k * 8 + c];
    s2 += zv * Wt2[k * 8 + c];
    s3 += zv * Wt3[k * 8 + c];
  }
  const bool mv = mask[row] != 0;
  t1[t] = mv ? s1 : 0.f;
  t2[t] = mv ? s2 : 0.f;
  t3[t] = mv ? s3 : 0.f;
}

// tg[b,0:8] = gf[b,:]@Wg ; mg[b,0:128] = gf[b,:]@Wmg
__global__ void graph_proj_kernel(const float* __restrict__ gf, const float* __restrict__ Wg,
                                  const float* __restrict__ Wmg,
                                  float* __restrict__ tg, float* __restrict__ mg) {
  const int t = blockIdx.x * blockDim.x + threadIdx.x;
  if (t >= 4 * 136) return;
  const int b = t / 136, c = t % 136;
  const float* g = gf + b * 128;
  if (c < 8) {
    float s = 0.f;
    for (int k = 0; k < 128; ++k) s += g[k] * Wg[k * 8 + c];
    tg[b * 8 + c] = s;
  } else {
    const int cc = c - 8;
    float s = 0.f;
    for (int k = 0; k < 128; ++k) s += g[k] * Wmg[k * 128 + cc];
    mg[b * 128 + cc] = s;
  }
}

// Wcat[128,32] = [We1 | We2 | We3 | 0]
__global__ void pack_wcat_kernel(const float* __restrict__ We1, const float* __restrict__ We2,
                                 const float* __restrict__ We3, float* __restrict__ Wcat) {
  const int t = blockIdx.x * blockDim.x + threadIdx.x;
  if (t >= 128 * 32) return;
  const int row = t >> 5, c = t & 31;
  float v = 0.f;
  if (c < 8)       v = We1[row * 8 + c];
  else if (c < 16) v = We2[row * 8 + (c - 8)];
  else if (c < 24) v = We3[row * 8 + (c - 16)];
  Wcat[t] = v;
}

// ---------------------------------------------------------------------------
// Triplet pooling: pooled[b,j,k,c] = max_{i valid} (t1[b,i,c]+e1[b,i,j,c]+e2[b,i,k,c])
//                                    + t2[b,j,c] + t3[b,k,c] + e3[b,j,k,c] + tg[b,c]
// valid(i) = mask[b,i] | mask[b,j] | mask[b,k]  (reference mask-OR semantics)
// bufT rows of 32: cols 0-7 = e1, 8-15 = e2, 16-23 = e3.
// ---------------------------------------------------------------------------
__global__ void tri_pool_kernel(const float* __restrict__ bufT, const float* __restrict__ t1,
                                const float* __restrict__ t2, const float* __restrict__ t3,
                                const float* __restrict__ tg,
                                const unsigned char* __restrict__ mask,
                                float* __restrict__ pooled) {
  const int bj = blockIdx.x;
  const int b = bj >> 7, j = bj & 127;
  const int k = threadIdx.x;
  const bool mjk = (mask[b * 128 + j] | mask[b * 128 + k]) != 0;
  float mx[8];
#pragma unroll
  for (int c = 0; c < 8; ++c) mx[c] = NEGINF;

  for (int i = 0; i < 128; ++i) {
    const bool use = mjk || (mask[b * 128 + i] != 0);
    if (!use) continue;
    const float* e1 = bufT + ((size_t)((b * 128 + i) * 128 + j)) * 32;
    const float* e2 = bufT + ((size_t)((b * 128 + i) * 128 + k)) * 32 + 8;
    const float* a  = t1 + (size_t)(b * 128 + i) * 8;
#pragma unroll
    for (int c = 0; c < 8; ++c) mx[c] = fmaxf(mx[c], a[c] + e1[c] + e2[c]);
  }
  const float* e3 = bufT + ((size_t)((b * 128 + j) * 128 + k)) * 32 + 16;
  float* outp = pooled + ((size_t)((b * 128 + j) * 128 + k)) * 8;
#pragma unroll
  for (int c = 0; c < 8; ++c)
    outp[c] = mx[c] + t2[(b * 128 + j) * 8 + c] + t3[(b * 128 + k) * 8 + c] + e3[c] +
              tg[b * 8 + c];
}

// tri_msgs[row,c] = relu(pooled[row,0:8] @ WU3[8,128])
__global__ void tri_out_kernel(const float* __restrict__ pooled, const float* __restrict__ WU3,
                               float* __restrict__ out) {
  const size_t r = blockIdx.x;
  const int c = threadIdx.x;
  const float* p = pooled + r * 8;
  float s = 0.f;
#pragma unroll
  for (int t = 0; t < 8; ++t) s += p[t] * WU3[t * 128 + c];
  out[r * 128 + c] = fmaxf(s, 0.f);
}

// ---------------------------------------------------------------------------
// Pairwise path elementwise kernels (wave-per-row LayerNorm, 128 channels)
// ---------------------------------------------------------------------------

// x = relu(LN( (adj? e@Wme:0) + (mask[j]? m1[b,j]:0) + (mask[i]? m2[b,i]:0) + mg[b] ))
__global__ void combine_ln1_kernel(const float* __restrict__ me, const float* __restrict__ m1,
                                   const float* __restrict__ m2, const float* __restrict__ mg,
                                   const unsigned char* __restrict__ adj,
                                   const unsigned char* __restrict__ mask,
                                   const float* __restrict__ s, const float* __restrict__ o,
                                   float* __restrict__ out) {
  const int row  = (int)((blockIdx.x * blockDim.x + threadIdx.x) >> 5);
  const int lane = threadIdx.x & 31;
  if (row >= 65536) return;
  const int b = row >> 14, ij = row & 16383, i = ij >> 7, j = ij & 127;
  const int c0 = lane * 4;

  float x[4] = {0.f, 0.f, 0.f, 0.f};
  if (adj[(size_t)b * 16384 + ij]) {
    const float* p = me + (size_t)row * 128 + c0;
#pragma unroll
    for (int t = 0; t < 4; ++t) x[t] = p[t];
  }
  if (mask[b * 128 + j]) {
    const float* p = m1 + (size_t)(b * 128 + j) * 128 + c0;
#pragma unroll
    for (int t = 0; t < 4; ++t) x[t] += p[t];
  }
  if (mask[b * 128 + i]) {
    const float* p = m2 + (size_t)(b * 128 + i) * 128 + c0;
#pragma unroll
    for (int t = 0; t < 4; ++t) x[t] += p[t];
  }
  {
    const float* p = mg + b * 128 + c0;
#pragma unroll
    for (int t = 0; t < 4; ++t) x[t] += p[t];
  }
  float sum = x[0] + x[1] + x[2] + x[3];
  sum = wave_sum32(sum);
  const float mean = sum * (1.f / 128.f);
  float d[4], sq = 0.f;
#pragma unroll
  for (int t = 0; t < 4; ++t) { d[t] = x[t] - mean; sq += d[t] * d[t]; }
  sq = wave_sum32(sq);
  const float inv = rsqrtf(sq * (1.f / 128.f) + LN_EPS);
  float* po = out + (size_t)row * 128 + c0;
#pragma unroll
  for (int t = 0; t < 4; ++t) po[t] = fmaxf(s[c0 + t] * inv * d[t] + o[c0 + t], 0.f);
}

// out = act(LN(in)) ; do_relu selects relu
__global__ void ln_act_kernel(const float* __restrict__ in, float* __restrict__ out,
                              const float* __restrict__ s, const float* __restrict__ o,
                              int nrows, int do_relu) {
  const int row  = (int)((blockIdx.x * blockDim.x + threadIdx.x) >> 5);
  const int lane = threadIdx.x & 31;
  if (row >= nrows) return;
  const int c0 = lane * 4;
  const float* p = in + (size_t)row * 128 + c0;
  float x[4];
#pragma unroll
  for (int t = 0; t < 4; ++t) x[t] = p[t];
  float sum = x[0] + x[1] + x[2] + x[3];
  sum = wave_sum32(sum);
  const float mean = sum * (1.f / 128.f);
  float d[4], sq = 0.f;
#pragma unroll
  for (int t = 0; t < 4; ++t) { d[t] = x[t] - mean; sq += d[t] * d[t]; }
  sq = wave_sum32(sq);
  const float inv = rsqrtf(sq * (1.f / 128.f) + LN_EPS);
  float* po = out + (size_t)row * 128 + c0;
#pragma unroll
  for (int t = 0; t < 4; ++t) {
    float v = s[c0 + t] * inv * d[t] + o[c0 + t];
    po[t] = do_relu ? fmaxf(v, 0.f) : v;
  }
}

// out = LN(a + b2)  (final ret, 512 rows)
__global__ void add2_ln_kernel(const float* __restrict__ a, const float* __restrict__ b2,
                               const float* __restrict__ s, const float* __restrict__ o,
                               float* __restrict__ out, int nrows) {
  const int row  = (int)((blockIdx.x * blockDim.x + threadIdx.x) >> 5);
  const int lane = threadIdx.x & 31;
  if (row >= nrows) return;
  const int c0 = lane * 4;
  const float* pa = a + (size_t)row * 128 + c0;
  const float* pb = b2 + (size_t)row * 128 + c0;
  float x[4];
#pragma unroll
  for (int t = 0; t < 4; ++t) x[t] = pa[t] + pb[t];
  float sum = x[0] + x[1] + x[2] + x[3];
  sum = wave_sum32(sum);
  const float mean = sum * (1.f / 128.f);
  float d[4], sq = 0.f;
#pragma unroll
  for (int t = 0; t < 4; ++t) { d[t] = x[t] - mean; sq += d[t] * d[t]; }
  sq = wave_sum32(sq);
  const float inv = rsqrtf(sq * (1.f / 128.f) + LN_EPS);
  float* po = out + (size_t)row * 128 + c0;
#pragma unroll
  for (int t = 0; t < 4; ++t) po[t] = s[c0 + t] * inv * d[t] + o[c0 + t];
}

// msgs_out[b,j,c] = max_i (adj[b,i,j] ? x[b,i,j,c] : NEG)
__global__ void masked_max_kernel(const float* __restrict__ x,
                                  const unsigned char* __restrict__ adj,
                                  float* __restrict__ out) {
  const int bj = blockIdx.x;           // 0..511
  const int b = bj >> 7, j = bj & 127;
  const int c = threadIdx.x;           // 0..127
  float best = NEGINF;
  const float* base = x + ((size_t)b * 16384 + j) * 128 + c;
  const unsigned char* arow = adj + (size_t)b * 16384 + j;
  for (int i = 0; i < 128; ++i) {
    if (arow[(size_t)i * 128]) best = fmaxf(best, base[(size_t)i * 128 * 128]);
  }
  out[(size_t)bj * 128 + c] = best;
}

// ---------------------------------------------------------------------------
// Host launch
// ---------------------------------------------------------------------------
extern "C" void kernel_launch(void* const* d_in, const int* in_sizes, int n_in,
                              void* d_out, int out_size, void* d_ws, size_t ws_size,
                              hipStream_t stream) {
  (void)in_sizes; (void)n_in; (void)out_size; (void)ws_size;
  const float* z    = (const float*)d_in[0];   // (4,128,256)
  const float* e2d  = (const float*)d_in[1];   // (4,128,128,128) -> (65536,128)
  const float* gf   = (const float*)d_in[2];   // (4,128)
  const unsigned char* mask = (const unsigned char*)d_in[3];  // (4,128) bool
  const unsigned char* adj  = (const unsigned char*)d_in[4];  // (4,128,128) bool
  const float* Wt1  = (const float*)d_in[5];
  const float* Wt2  = (const float*)d_in[6];
  const float* Wt3  = (const float*)d_in[7];
  const float* We1  = (const float*)d_in[8];
  const float* We2  = (const float*)d_in[9];
  const float* We3  = (const float*)d_in[10];
  const float* Wg   = (const float*)d_in[11];
  const float* Wm1  = (const float*)d_in[12];
  const float* Wm2  = (const float*)d_in[13];
  const float* Wme  = (const float*)d_in[14];
  const float* Wmg  = (const float*)d_in[15];
  const float* ln1s = (const float*)d_in[16];
  const float* ln1o = (const float*)d_in[17];
  const float* Wmlp1= (const float*)d_in[18];
  const float* ln2s = (const float*)d_in[19];
  const float* ln2o = (const float*)d_in[20];
  const float* Wmlp2= (const float*)d_in[21];
  const float* WU1  = (const float*)d_in[22];
  const float* WU2  = (const float*)d_in[23];
  const float* WU3  = (const float*)d_in[24];
  const float* lnfs = (const float*)d_in[25];
  const float* lnfo = (const float*)d_in[26];

  float* out = (float*)d_out;
  float* ret_out  = out;                 // (4,128,128)
  float* msgs_out = out + 65536;         // (4,128,128)
  float* tri_out  = out + 131072;        // (4,128,128,128)

  float* ws = (float*)d_ws;
  const size_t BIG = 65536ull * 128ull;  // 8,388,608 floats
  float* bufA = ws;                      // e@Wme, later LN2 output
  float* bufB = ws + BIG;                // x, later mlp2 output
  float* bufC = ws + 2 * BIG;            // bufT (e@Wcat), later mlp1 output
  float* sm   = ws + 3 * BIG;
  float* m1     = sm;               // 65536
  float* m2     = sm + 65536;       // 65536
  float* zU1    = sm + 131072;      // 65536
  float* mU2    = sm + 196608;      // 65536
  float* mg     = sm + 262144;      // 512
  float* t1     = sm + 262656;      // 4096
  float* t2     = sm + 266752;      // 4096
  float* t3     = sm + 270848;      // 4096
  float* tg     = sm + 274944;      // 32
  float* Wcat   = sm + 274976;      // 4096
  float* pooled = sm + 279072;      // 524288

  auto gemm = [&](const float* A, const float* W, float* C, int M, int K, int Nc) {
    dim3 grid(M / 64);
    dim3 block(128);
    size_t shmem = (size_t)K * Nc * sizeof(float);
    if (Nc == 128) wmma_gemm_lds<8><<<grid, block, shmem, stream>>>(A, W, C, M, K);
    else           wmma_gemm_lds<2><<<grid, block, shmem, stream>>>(A, W, C, M, K);
  };

  // ---- triplet path -------------------------------------------------------
  pack_wcat_kernel<<<16, 256, 0, stream>>>(We1, We2, We3, Wcat);
  z_proj_kernel<<<16, 256, 0, stream>>>(z, Wt1, Wt2, Wt3, mask, t1, t2, t3);
  graph_proj_kernel<<<3, 256, 0, stream>>>(gf, Wg, Wmg, tg, mg);
  gemm(e2d, Wcat, bufC, 65536, 128, 32);                 // bufT = [e1|e2|e3|0]
  tri_pool_kernel<<<512, 128, 0, stream>>>(bufC, t1, t2, t3, tg, mask, pooled);
  tri_out_kernel<<<65536, 128, 0, stream>>>(pooled, WU3, tri_out);

  // ---- pairwise path ------------------------------------------------------
  gemm(e2d, Wme, bufA, 65536, 128, 128);                 // msg_e (pre-adj)
  gemm(z, Wm1, m1, 512, 256, 128);
  gemm(z, Wm2, m2, 512, 256, 128);
  combine_ln1_kernel<<<8192, 256, 0, stream>>>(bufA, m1, m2, mg, adj, mask,
                                               ln1s, ln1o, bufB);
  gemm(bufB, Wmlp1, bufC, 65536, 128, 128);              // y = relu(ln1(msgs)) @ Wmlp1
  ln_act_kernel<<<8192, 256, 0, stream>>>(bufC, bufA, ln2s, ln2o, 65536, 1);
  gemm(bufA, Wmlp2, bufB, 65536, 128, 128);              // x2
  masked_max_kernel<<<512, 128, 0, stream>>>(bufB, adj, msgs_out);

  // ---- final readout ------------------------------------------------------
  gemm(z, WU1, zU1, 512, 256, 128);
  gemm(msgs_out, WU2, mU2, 512, 128, 128);
  add2_ln_kernel<<<64, 256, 0, stream>>>(zU1, mU2, lnfs, lnfo, ret_out, 512);
}